// HEATEncoder_38036230373818
// MI455X (gfx1250) — compile-verified
//
#include <hip/hip_runtime.h>
#include <cstdint>

typedef __attribute__((ext_vector_type(2))) float v2f;
typedef __attribute__((ext_vector_type(8))) float v8f;

#define HID    128
#define HEADS  4
#define EEMB   16
#define N_GEN  15000
#define N_BUS  50000
#define N_RES  5000
#define NTOT   (N_GEN + N_BUS + N_RES)
#define E_TOT  185000
#define SLOPE  0.2f

#define GB_GEN 118   // ceil(15000/128)
#define GB_BUS 391   // ceil(50000/128)
#define GB_RES 40    // ceil(5000/128)

__device__ __forceinline__ float lrelu_f(float v) { return v > 0.f ? v : SLOPE * v; }
// order-preserving float->uint key for atomicMax-based segment max
__device__ __forceinline__ unsigned fkey(float f) {
    unsigned u = __float_as_uint(f);
    return (u & 0x80000000u) ? ~u : (u | 0x80000000u);
}
__device__ __forceinline__ float funkey(unsigned k) {
    return (k & 0x80000000u) ? __uint_as_float(k ^ 0x80000000u) : __uint_as_float(~k);
}

// Fill LDS with a KROWS x 128 weight matrix pre-swizzled into WMMA B-fragment
// order: fragment (kk4, t), lane l holds the contiguous pair
//   ( W[kk4*4 + (l>>4)*2 + 0][t*16 + (l&15)], W[kk4*4 + (l>>4)*2 + 1][...] )
// at LDS dword index ((kk4*8 + t)*32 + l)*2.  One ds_load_b64 per fragment,
// bank-conflict-free (lane l -> banks 2l, 2l+1).
__device__ __forceinline__ void fill_bfrag(float* lds, const float* __restrict__ W,
                                           int krows, int tid, int nthreads) {
    int total = krows * HID;
    for (int i = tid; i < total; i += nthreads) {       // i = k*128 + c (coalesced global reads)
        int k = i >> 7, c = i & 127;
        int kk4 = k >> 2, kr = k & 3;
        int l = ((kr >> 1) << 4) + (c & 15);
        int idx = ((kk4 * 8 + (c >> 4)) * 32 + l) * 2 + (kr & 1);
        lds[idx] = W[i];
    }
}

// ---------------- input projections: x[N,128] ----------------
__global__ void proj_kernel(const float* __restrict__ xg, const float* __restrict__ xb,
                            const float* __restrict__ xr,
                            const float* __restrict__ wg, const float* __restrict__ bg,
                            const float* __restrict__ wb, const float* __restrict__ bb,
                            const float* __restrict__ wr, const float* __restrict__ br,
                            float* __restrict__ x) {
    int idx = blockIdx.x * blockDim.x + threadIdx.x;
    if (idx >= NTOT * HID) return;
    int n = idx >> 7, c = idx & 127;
    float acc;
    if (n < N_GEN) {
        acc = bg[c];
        const float* row = xg + n * 24;
        #pragma unroll
        for (int k = 0; k < 24; ++k) acc += row[k] * wg[k * HID + c];
    } else if (n < N_GEN + N_BUS) {
        acc = xb[n - N_GEN] * wb[c] + bb[c];
    } else {
        acc = xr[n - N_GEN - N_BUS] * wr[c] + br[c];
    }
    x[idx] = acc;
}

// ---------------- HeteroLinear h = x @ W[type] + b, f32 WMMA ----------------
__global__ void __launch_bounds__(256) hetero_lin_kernel(const float* __restrict__ x,
                                                         const float* __restrict__ hlin_w,
                                                         const float* __restrict__ hlin_b,
                                                         float* __restrict__ h) {
    __shared__ __align__(16) float lw[HID * HID];   // B-fragment swizzled
    __shared__ float lb[HID];
    int bid = blockIdx.x;
    int type, r0, segEnd;
    if (bid < GB_GEN)                 { type = 0; r0 = bid * 128;                              segEnd = N_GEN; }
    else if (bid < GB_GEN + GB_BUS)   { type = 1; r0 = N_GEN + (bid - GB_GEN) * 128;           segEnd = N_GEN + N_BUS; }
    else                              { type = 2; r0 = N_GEN + N_BUS + (bid - GB_GEN - GB_BUS) * 128; segEnd = NTOT; }

    fill_bfrag(lw, hlin_w + (size_t)type * HID * HID, HID, threadIdx.x, 256);
    if (threadIdx.x < HID) lb[threadIdx.x] = hlin_b[type * HID + threadIdx.x];
    __syncthreads();

    int wave = threadIdx.x >> 5;
    int lane = threadIdx.x & 31;
    int hv   = lane >> 4;
    int l16  = lane & 15;
    int rw   = r0 + wave * 16;

    v8f acc[8];
    #pragma unroll
    for (int t = 0; t < 8; ++t)
        #pragma unroll
        for (int i = 0; i < 8; ++i) acc[t][i] = 0.f;

    int arow = rw + l16; if (arow >= segEnd) arow = segEnd - 1;
    const float* xrow = x + (size_t)arow * HID;
    const v2f* lwv = (const v2f*)lw;

    for (int kk4 = 0; kk4 < 32; ++kk4) {
        v2f a = *(const v2f*)(xrow + (kk4 << 2) + hv * 2);
        #pragma unroll
        for (int t = 0; t < 8; ++t) {
            v2f b = lwv[(kk4 * 8 + t) * 32 + lane];
            acc[t] = __builtin_amdgcn_wmma_f32_16x16x4_f32(false, a, false, b,
                                                           (short)0, acc[t], false, false);
        }
    }

    if (r0 + 128 <= segEnd) {           // full-tile fast path: no per-element guards
        #pragma unroll
        for (int t = 0; t < 8; ++t) {
            int col = t * 16 + l16;
            float bias = lb[col];
            #pragma unroll
            for (int r = 0; r < 8; ++r)
                h[(size_t)(rw + r + hv * 8) * HID + col] = acc[t][r] + bias;
        }
    } else {
        #pragma unroll
        for (int t = 0; t < 8; ++t) {
            int col = t * 16 + l16;
            float bias = lb[col];
            #pragma unroll
            for (int r = 0; r < 8; ++r) {
                int row = rw + r + hv * 8;
                if (row < segEnd) h[(size_t)row * HID + col] = acc[t][r] + bias;
            }
        }
    }
}

// ---------------- node-level attention projections s[N,8] ----------------
__global__ void att_node_kernel(const float* __restrict__ h, const float* __restrict__ att_w,
                                float* __restrict__ s) {
    int idx = blockIdx.x * blockDim.x + threadIdx.x;
    if (idx >= NTOT * 8) return;
    int n = idx >> 3, j = idx & 7;
    int head = j & 3;
    int rbase = (j < 4) ? 0 : HID;
    const float* hrow = h + (size_t)n * HID;
    float acc = 0.f;
    #pragma unroll 4
    for (int k = 0; k < HID; ++k) acc += hrow[k] * att_w[(rbase + k) * HEADS + head];
    s[idx] = acc;
}

// ---------------- per-edge-type attention constant ----------------
__global__ void et_term_kernel(const float* __restrict__ etype_emb,
                               const float* __restrict__ att_w, float* __restrict__ et_term) {
    if (threadIdx.x >= 16) return;
    int t = threadIdx.x >> 2, hd = threadIdx.x & 3;
    float acc = 0.f;
    for (int k = 0; k < EEMB; ++k)
        acc += lrelu_f(etype_emb[t * EEMB + k]) * att_w[(256 + k) * HEADS + hd];
    et_term[t * 4 + hd] = acc;
}

// ---------------- edge pass 1: ea_emb, raw alpha, segment max ----------------
__global__ void edge_alpha_kernel(const int* __restrict__ ei, const float* __restrict__ ea,
                                  int count, int etype, int eoff,
                                  const float* __restrict__ s, const float* __restrict__ et_term,
                                  const float* __restrict__ eattr_w, const float* __restrict__ att_w,
                                  float* __restrict__ eaemb, float* __restrict__ exbuf,
                                  unsigned* __restrict__ amaxkey) {
    int e = blockIdx.x * blockDim.x + threadIdx.x;
    if (e >= count) return;
    int src = ei[e], dst = ei[count + e];
    float a0 = ea[(size_t)e * 2], a1 = ea[(size_t)e * 2 + 1];
    float em[EEMB];
    #pragma unroll
    for (int k = 0; k < EEMB; ++k) em[k] = lrelu_f(a0 * eattr_w[k] + a1 * eattr_w[EEMB + k]);
    float4* eo = (float4*)(eaemb + (size_t)(eoff + e) * EEMB);
    eo[0] = make_float4(em[0], em[1], em[2], em[3]);
    eo[1] = make_float4(em[4], em[5], em[6], em[7]);
    eo[2] = make_float4(em[8], em[9], em[10], em[11]);
    eo[3] = make_float4(em[12], em[13], em[14], em[15]);
    #pragma unroll
    for (int hd = 0; hd < 4; ++hd) {
        float al = s[(size_t)dst * 8 + hd] + s[(size_t)src * 8 + 4 + hd] + et_term[etype * 4 + hd];
        #pragma unroll
        for (int k = 0; k < EEMB; ++k) al += em[k] * att_w[(272 + k) * HEADS + hd];
        al = lrelu_f(al);
        exbuf[(size_t)(eoff + e) * 4 + hd] = al;
        atomicMax(&amaxkey[(size_t)dst * 4 + hd], fkey(al));
    }
}

// ---------------- edge pass 2: exp + segment sum ----------------
__global__ void edge_exp_kernel(const int* __restrict__ ei, int count, int eoff,
                                const unsigned* __restrict__ amaxkey,
                                float* __restrict__ exbuf, float* __restrict__ den) {
    int e = blockIdx.x * blockDim.x + threadIdx.x;
    if (e >= count) return;
    int dst = ei[count + e];
    #pragma unroll
    for (int hd = 0; hd < 4; ++hd) {
        float am = funkey(amaxkey[(size_t)dst * 4 + hd]);
        float ex = __expf(exbuf[(size_t)(eoff + e) * 4 + hd] - am);
        exbuf[(size_t)(eoff + e) * 4 + hd] = ex;
        atomicAdd(&den[(size_t)dst * 4 + hd], ex);
    }
}

// ---------------- edge pass 3: per-edge scalar weight (head mean) ----------------
__global__ void edge_w_kernel(const int* __restrict__ ei, int count, int eoff,
                              const float* __restrict__ exbuf, const float* __restrict__ den,
                              float* __restrict__ wbuf) {
    int e = blockIdx.x * blockDim.x + threadIdx.x;
    if (e >= count) return;
    int dst = ei[count + e];
    float acc = 0.f;
    #pragma unroll
    for (int hd = 0; hd < 4; ++hd)
        acc += exbuf[(size_t)(eoff + e) * 4 + hd] / (den[(size_t)dst * 4 + hd] + 1e-16f);
    wbuf[eoff + e] = 0.25f * acc;
}

// ---------------- edge pass 4: message GEMM (K=144) + weighted scatter, f32 WMMA ----------------
__global__ void __launch_bounds__(256) edge_msg_kernel(const int* __restrict__ ei, int count, int eoff,
                                                       const float* __restrict__ h,
                                                       const float* __restrict__ eaemb,
                                                       const float* __restrict__ lin_w,
                                                       const float* __restrict__ wbuf,
                                                       float* __restrict__ agg) {
    __shared__ __align__(16) float lw[144 * HID];   // B-fragment swizzled
    fill_bfrag(lw, lin_w, 144, threadIdx.x, 256);
    __syncthreads();

    int wave = threadIdx.x >> 5;
    int lane = threadIdx.x & 31;
    int hv   = lane >> 4;
    int l16  = lane & 15;
    int e0   = (blockIdx.x * 8 + wave) * 16;
    if (e0 >= count) return;

    int rowe = e0 + l16;
    int rowc = rowe < count ? rowe : count - 1;
    int srcn = ei[rowc];
    const float* hrow  = h + (size_t)srcn * HID;
    const float* earow = eaemb + (size_t)(eoff + rowc) * EEMB;
    const v2f* lwv = (const v2f*)lw;

    v8f acc[8];
    #pragma unroll
    for (int t = 0; t < 8; ++t)
        #pragma unroll
        for (int i = 0; i < 8; ++i) acc[t][i] = 0.f;

    for (int kk4 = 0; kk4 < 32; ++kk4) {            // K = 0..127 from h[src]
        v2f a = *(const v2f*)(hrow + (kk4 << 2) + hv * 2);
        #pragma unroll
        for (int t = 0; t < 8; ++t) {
            v2f b = lwv[(kk4 * 8 + t) * 32 + lane];
            acc[t] = __builtin_amdgcn_wmma_f32_16x16x4_f32(false, a, false, b,
                                                           (short)0, acc[t], false, false);
        }
    }
    #pragma unroll
    for (int kk4 = 32; kk4 < 36; ++kk4) {           // K = 128..143 from ea_emb
        v2f a = *(const v2f*)(earow + ((kk4 - 32) << 2) + hv * 2);
        #pragma unroll
        for (int t = 0; t < 8; ++t) {
            v2f b = lwv[(kk4 * 8 + t) * 32 + lane];
            acc[t] = __builtin_amdgcn_wmma_f32_16x16x4_f32(false, a, false, b,
                                                           (short)0, acc[t], false, false);
        }
    }

    if (e0 + 16 <= count) {             // full-tile fast path
        int   dste[8];
        float we[8];
        #pragma unroll
        for (int r = 0; r < 8; ++r) {
            int e = e0 + r + hv * 8;
            dste[r] = ei[count + e];
            we[r]   = wbuf[eoff + e];
        }
        #pragma unroll
        for (int t = 0; t < 8; ++t) {
            int col = t * 16 + l16;
            #pragma unroll
            for (int r = 0; r < 8; ++r)
                atomicAdd(&agg[(size_t)dste[r] * HID + col], acc[t][r] * we[r]);
        }
    } else {
        int   dste[8];
        float we[8];
        #pragma unroll
        for (int r = 0; r < 8; ++r) {
            int e = e0 + r + hv * 8;
            if (e < count) { dste[r] = ei[count + e]; we[r] = wbuf[eoff + e]; }
            else           { dste[r] = -1;            we[r] = 0.f; }
        }
        #pragma unroll
        for (int t = 0; t < 8; ++t) {
            int col = t * 16 + l16;
            #pragma unroll
            for (int r = 0; r < 8; ++r)
                if (dste[r] >= 0)
                    atomicAdd(&agg[(size_t)dste[r] * HID + col], acc[t][r] * we[r]);
        }
    }
}

// ---------------- epilogues ----------------
__global__ void relu_kernel(const float* __restrict__ a, float* __restrict__ o, int n) {
    int i = blockIdx.x * blockDim.x + threadIdx.x;
    if (i < n) o[i] = fmaxf(a[i], 0.f);
}
__global__ void final_kernel(const float* __restrict__ agg, const float* __restrict__ x1,
                             float* __restrict__ out, int n) {
    int i = blockIdx.x * blockDim.x + threadIdx.x;
    if (i < n) out[i] = fmaxf(agg[i], 0.f) + x1[i];
}

// ---------------- host orchestration ----------------
extern "C" void kernel_launch(void* const* d_in, const int* in_sizes, int n_in,
                              void* d_out, int out_size, void* d_ws, size_t ws_size,
                              hipStream_t stream) {
    (void)in_sizes; (void)n_in; (void)out_size; (void)ws_size;
    const float* xg      = (const float*)d_in[0];
    const float* xb      = (const float*)d_in[1];
    const float* xr      = (const float*)d_in[2];
    const int*   ei_prod = (const int*)d_in[3];
    const float* ea_prod = (const float*)d_in[4];
    const int*   ei_serv = (const int*)d_in[5];
    const float* ea_serv = (const float*)d_in[6];
    const int*   ei_tran = (const int*)d_in[7];
    const float* ea_tran = (const float*)d_in[8];
    const int*   ei_back = (const int*)d_in[9];
    const float* ea_back = (const float*)d_in[10];
    const float* p_gen_w = (const float*)d_in[11];
    const float* p_gen_b = (const float*)d_in[12];
    const float* p_bus_w = (const float*)d_in[13];
    const float* p_bus_b = (const float*)d_in[14];
    const float* p_res_w = (const float*)d_in[15];
    const float* p_res_b = (const float*)d_in[16];
    // heat1: 17..22, heat2: 23..28  (hlin_w, hlin_b, etype_emb, eattr_w, att_w, lin_w)
    const float* heat[2][6];
    for (int l = 0; l < 2; ++l)
        for (int j = 0; j < 6; ++j)
            heat[l][j] = (const float*)d_in[17 + l * 6 + j];

    char* base = (char*)d_ws;
    size_t off = 0;
    auto alloc = [&](size_t nElem) { void* p = base + off; off += nElem * 4; return p; };
    float*    x       = (float*)alloc((size_t)NTOT * HID);
    float*    hbuf    = (float*)alloc((size_t)NTOT * HID);
    float*    s       = (float*)alloc((size_t)NTOT * 8);
    float*    eaemb   = (float*)alloc((size_t)E_TOT * EEMB);
    float*    exbuf   = (float*)alloc((size_t)E_TOT * 4);
    float*    wbuf    = (float*)alloc((size_t)E_TOT);
    float*    agg     = (float*)alloc((size_t)NTOT * HID);
    float*    x1      = (float*)alloc((size_t)NTOT * HID);
    float*    den     = (float*)alloc((size_t)NTOT * 4);
    unsigned* amaxkey = (unsigned*)alloc((size_t)NTOT * 4);
    float*    etterm  = (float*)alloc(16);

    struct ES { const int* ei; const float* ea; int cnt; int etype; int eoff; };
    ES es[4] = { { ei_prod, ea_prod, 15000,  0, 0      },
                 { ei_serv, ea_serv, 15000,  1, 15000  },
                 { ei_tran, ea_tran, 150000, 2, 30000  },
                 { ei_back, ea_back, 5000,   3, 180000 } };

    proj_kernel<<<(NTOT * HID + 255) / 256, 256, 0, stream>>>(
        xg, xb, xr, p_gen_w, p_gen_b, p_bus_w, p_bus_b, p_res_w, p_res_b, x);

    for (int layer = 0; layer < 2; ++layer) {
        const float* hlin_w    = heat[layer][0];
        const float* hlin_b    = heat[layer][1];
        const float* etype_emb = heat[layer][2];
        const float* eattr_w   = heat[layer][3];
        const float* att_w     = heat[layer][4];
        const float* lin_w     = heat[layer][5];
        const float* xin = (layer == 0) ? x : x1;

        hipMemsetAsync(agg, 0, (size_t)NTOT * HID * 4, stream);
        hipMemsetAsync(amaxkey, 0, (size_t)NTOT * 4 * 4, stream);
        hipMemsetAsync(den, 0, (size_t)NTOT * 4 * 4, stream);

        et_term_kernel<<<1, 16, 0, stream>>>(etype_emb, att_w, etterm);
        hetero_lin_kernel<<<GB_GEN + GB_BUS + GB_RES, 256, 0, stream>>>(xin, hlin_w, hlin_b, hbuf);
        att_node_kernel<<<(NTOT * 8 + 255) / 256, 256, 0, stream>>>(hbuf, att_w, s);

        for (int i = 0; i < 4; ++i)
            edge_alpha_kernel<<<(es[i].cnt + 255) / 256, 256, 0, stream>>>(
                es[i].ei, es[i].ea, es[i].cnt, es[i].etype, es[i].eoff,
                s, etterm, eattr_w, att_w, eaemb, exbuf, amaxkey);
        for (int i = 0; i < 4; ++i)
            edge_exp_kernel<<<(es[i].cnt + 255) / 256, 256, 0, stream>>>(
                es[i].ei, es[i].cnt, es[i].eoff, amaxkey, exbuf, den);
        for (int i = 0; i < 4; ++i)
            edge_w_kernel<<<(es[i].cnt + 255) / 256, 256, 0, stream>>>(
                es[i].ei, es[i].cnt, es[i].eoff, exbuf, den, wbuf);
        for (int i = 0; i < 4; ++i)
            edge_msg_kernel<<<(es[i].cnt + 127) / 128, 256, 0, stream>>>(
                es[i].ei, es[i].cnt, es[i].eoff, hbuf, eaemb, lin_w, wbuf, agg);

        if (layer == 0)
            relu_kernel<<<(NTOT * HID + 255) / 256, 256, 0, stream>>>(agg, x1, NTOT * HID);
        else
            final_kernel<<<(NTOT * HID + 255) / 256, 256, 0, stream>>>(agg, x1, (float*)d_out, NTOT * HID);
    }
}